// SphericalMHA_53188874994215
// MI455X (gfx1250) — compile-verified
//
#include <hip/hip_runtime.h>

#define DIM    768
#define NHEADS 12
#define HDIM   64
#define BATCH  8
#define SEQ    1024
#define SCALEF 0.125f   // HEAD_DIM^-0.5 = 64^-0.5

typedef __attribute__((ext_vector_type(2))) float v2f;
typedef __attribute__((ext_vector_type(8))) float v8f;

// D = A(16x4 f32) * B(4x16 f32) + C(16x16 f32), wave32 WMMA
__device__ __forceinline__ v8f wmma4(v2f a, v2f b, v8f c) {
  return __builtin_amdgcn_wmma_f32_16x16x4_f32(
      false, a, false, b, (short)0, c, false, false);
}

__device__ __forceinline__ v2f ld2(const float* p) {
  const float2 t = *reinterpret_cast<const float2*>(p);
  v2f r; r.x = t.x; r.y = t.y; return r;
}

// CDNA5 async global->LDS copy, 16B per lane, tracked by ASYNCcnt.
__device__ __forceinline__ void async_copy16(unsigned lds_byte, const float* g) {
  asm volatile("global_load_async_to_lds_b128 %0, %1, off"
               :: "v"(lds_byte), "v"((unsigned long long)(uintptr_t)g)
               : "memory");
}
#define WAIT_ASYNC_0() asm volatile("s_wait_asynccnt 0x0" ::: "memory")
#define WAIT_ASYNC_2() asm volatile("s_wait_asynccnt 0x2" ::: "memory")
#define WAIT_ASYNC_4() asm volatile("s_wait_asynccnt 0x4" ::: "memory")

__device__ __forceinline__ unsigned lds_addr(const void* p) {
  return (unsigned)(uintptr_t)p;   // low 32 bits of generic LDS ptr = DS offset
}

// ---------------------------------------------------------------------------
// LDS-tiled GEMM: Y = X @ W^T + bias.  Block = 8 waves, 128(M) x 128(N) tile,
// K staged in 16-deep double-buffered LDS tiles via async-to-LDS DMA.
// QKV=true scatters the 2304 output channels into head-major q/k/v buffers.
// ---------------------------------------------------------------------------
template <bool QKV>
__global__ __launch_bounds__(256) void gemm128_kernel(
    const float* __restrict__ X, const float* __restrict__ W,
    const float* __restrict__ bias, float* __restrict__ out0,
    float* __restrict__ out1, float* __restrict__ out2, int nc) {
  __shared__ float Xs[2][128 * 20];   // 20-float row stride: 16B-aligned rows,
  __shared__ float Ws[2][128 * 20];   // conflict-free b64 fragment reads

  const int t    = threadIdx.x;
  const int lane = t & 31;
  const int wave = t >> 5;
  const int mb   = blockIdx.x / nc;
  const int cb   = blockIdx.x % nc;
  const int m0   = mb * 128;
  const int c0   = cb * 128;
  const int r    = lane & 15;
  const int kh   = (lane >> 4) << 1;        // 0 or 2

  // staging map: thread -> (row, 4-float chunk); 2 rows apart by 64
  const int srow  = t >> 2;                 // 0..63
  const int spart = (t & 3) << 2;           // 0,4,8,12 (floats)

  const float* Xg = X + (size_t)(m0 + srow) * DIM + spart;
  const float* Wg = W + (size_t)(c0 + srow) * DIM + spart;

  v8f acc[8] = {};

  auto stage = [&](int buf, int k0) {
    async_copy16(lds_addr(&Xs[buf][srow * 20 + spart]),        Xg + k0);
    async_copy16(lds_addr(&Xs[buf][(srow + 64) * 20 + spart]), Xg + (size_t)64 * DIM + k0);
    async_copy16(lds_addr(&Ws[buf][srow * 20 + spart]),        Wg + k0);
    async_copy16(lds_addr(&Ws[buf][(srow + 64) * 20 + spart]), Wg + (size_t)64 * DIM + k0);
  };

  stage(0, 0);
  int it = 0;
  for (int k0 = 0; k0 < DIM; k0 += 16, ++it) {
    const int cur = it & 1;
    if (k0 + 16 < DIM) { stage(cur ^ 1, k0 + 16); WAIT_ASYNC_4(); }
    else               { WAIT_ASYNC_0(); }
    __syncthreads();

    const float* xb = &Xs[cur][0];
    const float* wb = &Ws[cur][0];
    #pragma unroll
    for (int kk = 0; kk < 4; ++kk) {
      const int c = kk * 4 + kh;
      v2f a = ld2(&xb[(wave * 16 + r) * 20 + c]);
      #pragma unroll
      for (int ct = 0; ct < 8; ++ct) {
        v2f b = ld2(&wb[(ct * 16 + r) * 20 + c]);
        acc[ct] = wmma4(a, b, acc[ct]);
      }
    }
    __syncthreads();   // all waves done with buf[cur] before it is re-staged
  }

  const int rowbase = m0 + wave * 16 + ((lane >> 4) << 3);
  #pragma unroll
  for (int ct = 0; ct < 8; ++ct) {
    const int o = c0 + ct * 16 + r;
    const float bv = bias[o];
    if (QKV) {
      const int s   = o / DIM;              // 0=q, 1=k, 2=v
      const int rem = o - s * DIM;
      const int h   = rem >> 6;
      const int d   = rem & 63;
      float* dst = (s == 0) ? out0 : ((s == 1) ? out1 : out2);
      #pragma unroll
      for (int i = 0; i < 8; ++i) {
        const int m = rowbase + i;
        const int b = m >> 10;
        const int n = m & 1023;
        dst[(((size_t)b * NHEADS + h) * SEQ + n) * HDIM + d] = acc[ct][i] + bv;
      }
    } else {
      #pragma unroll
      for (int i = 0; i < 8; ++i)
        out0[(size_t)(rowbase + i) * DIM + o] = acc[ct][i] + bv;
    }
  }
}

// ---------------------------------------------------------------------------
// Flash attention: 8 waves per block share one (b,h); each wave owns a
// 16-query block. K/V 16-key tiles are async-staged into double-buffered LDS
// (read once per 128 queries instead of once per 16). Streaming softmax kept
// in WMMA C-layout; P transposed to A-layout through per-wave LDS.
// ---------------------------------------------------------------------------
__global__ __launch_bounds__(256) void attn_kernel(
    const float* __restrict__ Qb, const float* __restrict__ Kb,
    const float* __restrict__ Vb, const float* __restrict__ lat_bias,
    const int* __restrict__ mask, float* __restrict__ AO) {
  __shared__ float kt[2][16 * 68];    // 68-float stride: 16B rows, bank-spread
  __shared__ float vt[2][16 * 68];
  __shared__ float pstage[8][16][17];

  const int t    = threadIdx.x;
  const int lane = t & 31;
  const int wave = t >> 5;
  const int wid  = blockIdx.x;                  // 0 .. B*H*8-1
  const int qg   = wid & 7;                     // 128-query group
  const int bh   = wid >> 3;
  const int h    = bh % NHEADS;
  const int b    = bh / NHEADS;
  const float* Q = Qb + (size_t)bh * SEQ * HDIM;
  const float* K = Kb + (size_t)bh * SEQ * HDIM;
  const float* V = Vb + (size_t)bh * SEQ * HDIM;
  const int r     = lane & 15;
  const int kh    = (lane >> 4) << 1;
  const int half8 = (lane >> 4) << 3;
  const int q0    = qg * 128 + wave * 16;

  // staging map: 256 threads x 16B cover one 16x64 tile
  const int srow  = t >> 4;                     // 0..15
  const int spart = (t & 15) << 2;              // 0..60 floats

  auto stage = [&](int buf, int j) {
    async_copy16(lds_addr(&kt[buf][srow * 68 + spart]), K + (size_t)(j + srow) * HDIM + spart);
    async_copy16(lds_addr(&vt[buf][srow * 68 + spart]), V + (size_t)(j + srow) * HDIM + spart);
  };

  // Q A-fragments for all 16 K-steps over head dim (reused every key block)
  v2f aq[16];
  #pragma unroll
  for (int dk = 0; dk < 16; ++dk)
    aq[dk] = ld2(Q + (size_t)(q0 + r) * HDIM + dk * 4 + kh);

  v8f o0 = {}, o1 = {}, o2 = {}, o3 = {};
  float mst[8], lst[8];
  #pragma unroll
  for (int i = 0; i < 8; ++i) { mst[i] = -3.0e38f; lst[i] = 0.0f; }

  stage(0, 0);
  int it = 0;
  for (int j = 0; j < SEQ; j += 16, ++it) {
    const int cur = it & 1;
    if (j + 16 < SEQ) { stage(cur ^ 1, j + 16); WAIT_ASYNC_2(); }
    else              { WAIT_ASYNC_0(); }
    __syncthreads();

    // S = Q @ K^T from the staged K tile
    v8f s = {};
    #pragma unroll
    for (int dk = 0; dk < 16; ++dk) {
      v2f bk = ld2(&kt[cur][r * 68 + dk * 4 + kh]);
      s = wmma4(aq[dk], bk, s);
    }

    // scale + per-key bias + mask (C-layout column == lane&15)
    const float cb = lat_bias[(size_t)b * SEQ + j + r];
    const int* mrow = mask + (size_t)b * SEQ * SEQ + (size_t)(q0 + half8) * SEQ + j + r;
    #pragma unroll
    for (int i = 0; i < 8; ++i) {
      float sv = s[i] * SCALEF + cb;
      s[i] = (mrow[(size_t)i * SEQ] == 0) ? -3.0e38f : sv;
    }

    // online softmax: reductions across 16-lane halves; row ids align with
    // the O accumulator layout so the rescale is purely lane-local
    #pragma unroll
    for (int i = 0; i < 8; ++i) {
      float rm = s[i];
      rm = fmaxf(rm, __shfl_xor(rm, 1, 32));
      rm = fmaxf(rm, __shfl_xor(rm, 2, 32));
      rm = fmaxf(rm, __shfl_xor(rm, 4, 32));
      rm = fmaxf(rm, __shfl_xor(rm, 8, 32));
      const float mnew  = fmaxf(mst[i], rm);
      const float alpha = __expf(mst[i] - mnew);
      mst[i] = mnew;
      const float p = __expf(s[i] - mnew);
      s[i] = p;
      float rs = p;
      rs += __shfl_xor(rs, 1, 32);
      rs += __shfl_xor(rs, 2, 32);
      rs += __shfl_xor(rs, 4, 32);
      rs += __shfl_xor(rs, 8, 32);
      lst[i] = lst[i] * alpha + rs;
      o0[i] *= alpha; o1[i] *= alpha; o2[i] *= alpha; o3[i] *= alpha;
    }

    // P: C-layout -> A-layout via per-wave LDS (DS ops in-order within a wave)
    #pragma unroll
    for (int i = 0; i < 8; ++i) pstage[wave][half8 + i][r] = s[i];

    // O += P @ V from the staged V tile
    #pragma unroll
    for (int kk = 0; kk < 16; kk += 4) {
      v2f ap; ap.x = pstage[wave][r][kk + kh]; ap.y = pstage[wave][r][kk + kh + 1];
      const float* vbase = &vt[cur][(kk + kh) * 68 + r];
      v2f bv;
      bv.x = vbase[0];  bv.y = vbase[68 + 0];  o0 = wmma4(ap, bv, o0);
      bv.x = vbase[16]; bv.y = vbase[68 + 16]; o1 = wmma4(ap, bv, o1);
      bv.x = vbase[32]; bv.y = vbase[68 + 32]; o2 = wmma4(ap, bv, o2);
      bv.x = vbase[48]; bv.y = vbase[68 + 48]; o3 = wmma4(ap, bv, o3);
    }
    __syncthreads();   // everyone done with buf[cur] before it is re-staged
  }

  // epilogue: normalize and write (b, n, h*64+d) layout for the proj GEMM
  #pragma unroll
  for (int i = 0; i < 8; ++i) {
    const float inv = 1.0f / lst[i];
    const int qrow = q0 + half8 + i;
    float* dst = AO + ((size_t)b * SEQ + qrow) * DIM + h * HDIM + r;
    dst[0]  = o0[i] * inv;
    dst[16] = o1[i] * inv;
    dst[32] = o2[i] * inv;
    dst[48] = o3[i] * inv;
  }
}

// ---------------------------------------------------------------------------
extern "C" void kernel_launch(void* const* d_in, const int* in_sizes, int n_in,
                              void* d_out, int out_size, void* d_ws, size_t ws_size,
                              hipStream_t stream) {
  (void)in_sizes; (void)n_in; (void)out_size; (void)ws_size;
  const float* x        = (const float*)d_in[0];
  const float* lat_bias = (const float*)d_in[1];
  const int*   mask     = (const int*)d_in[2];
  const float* qkv_w    = (const float*)d_in[3];
  const float* qkv_b    = (const float*)d_in[4];
  const float* proj_w   = (const float*)d_in[5];
  const float* proj_b   = (const float*)d_in[6];
  float* out = (float*)d_out;

  // workspace: q, k, v (head-major) + attention output, ~100 MB (L2-resident)
  const size_t plane = (size_t)BATCH * SEQ * DIM;   // 8192*768 floats
  float* ws = (float*)d_ws;
  float* qb = ws;
  float* kb = qb + plane;
  float* vb = kb + plane;
  float* ao = vb + plane;

  // QKV GEMM: (8192/128) x (2304/128) = 64 x 18 blocks
  gemm128_kernel<true><<<64 * 18, 256, 0, stream>>>(x, qkv_w, qkv_b, qb, kb, vb, 18);
  // Attention: B*H * (1024/128) = 96 * 8 blocks, 8 waves each
  attn_kernel<<<BATCH * NHEADS * 8, 256, 0, stream>>>(qb, kb, vb, lat_bias, mask, ao);
  // Proj GEMM: (8192/128) x (768/128) = 64 x 6 blocks
  gemm128_kernel<false><<<64 * 6, 256, 0, stream>>>(ao, proj_w, proj_b, out, nullptr, nullptr, 6);
}